// InnerLSTM_3994319585338
// MI455X (gfx1250) — compile-verified
//
#include <hip/hip_runtime.h>
#include <math.h>

#define VOCAB 50257
#define EMB   512
#define HID   512
#define SEQ   4096
#define G4    2048  // 4*HID

typedef __attribute__((ext_vector_type(2))) float v2f;
typedef __attribute__((ext_vector_type(8))) float v8f;
typedef int v4i __attribute__((__vector_size__(16)));

#define AS1 __attribute__((address_space(1)))
#define AS3 __attribute__((address_space(3)))

// Async global->LDS path (GLOBAL_LOAD_ASYNC_TO_LDS_*, s_wait_asynccnt) if the
// toolchain exposes the builtins; otherwise fall back to VGPR staging.
#if defined(__has_builtin)
#if __has_builtin(__builtin_amdgcn_global_load_async_to_lds_b128) && \
    __has_builtin(__builtin_amdgcn_global_load_async_to_lds_b32)  && \
    __has_builtin(__builtin_amdgcn_s_wait_asynccnt)
#define USE_ASYNC_LDS 1
#endif
#endif
#define CPOL_SCOPE_DEV 0x10   // gfx12 CPol: scope field bits[4:3] = 2 (DEV)

// ---------------------------------------------------------------------------
// Init: zero the h double-buffer and the global barrier counter (deterministic
// across graph replays).
// ---------------------------------------------------------------------------
__global__ void init_state(float* hbuf, int* cnt) {
    int t = threadIdx.x;
    for (int i = t; i < 2 * HID; i += blockDim.x) hbuf[i] = 0.0f;
    if (t == 0) *cnt = 0;
}

// ---------------------------------------------------------------------------
// Phase 1: gx[s, r] = sum_e emb[tokens[s], e] * W_ih[r, e] + b_ih[r] + b_hh[r]
// One 16x16 output tile per wave via V_WMMA_F32_16X16X4_F32 (f32 exact).
// Block = 256 threads (8 waves) = one 16-row s-tile x 8 r-tiles.
// X tile staged into LDS via async-to-LDS DMA (no VGPR round trip).
// ---------------------------------------------------------------------------
__global__ void __launch_bounds__(256)
gx_gemm(const int* __restrict__ tokens, const float* __restrict__ emb,
        const float* __restrict__ Wih, const float* __restrict__ bih,
        const float* __restrict__ bhh, float* __restrict__ gx) {
    __shared__ __align__(16) float xs[16 * EMB];  // 32 KB gathered X tile
    __shared__ int stok[16];

    const int tid = threadIdx.x;
    const int s0  = blockIdx.x * 16;

    if (tid < 16) stok[tid] = tokens[s0 + tid];
    __syncthreads();

#ifdef USE_ASYNC_LDS
    // 2048 float4's: 8 async b128 copies per thread, straight into LDS.
    for (int i4 = tid; i4 < (16 * EMB) / 4; i4 += 256) {
        int row = i4 >> 7;              // i4 / (EMB/4)
        int col = (i4 & 127) << 2;
        const float* src = emb + (size_t)stok[row] * EMB + col;
        __builtin_amdgcn_global_load_async_to_lds_b128(
            (AS1 v4i*)src, (AS3 v4i*)(xs + (i4 << 2)), 0, 0);
    }
    __builtin_amdgcn_s_wait_asynccnt(0);
    __syncthreads();
#else
    for (int i = tid; i < 16 * EMB; i += 256) {
        int row = i >> 9, col = i & (EMB - 1);
        xs[i] = emb[(size_t)stok[row] * EMB + col];
    }
    __syncthreads();
#endif

    const int wave  = tid >> 5;
    const int lane  = tid & 31;
    const int rbase = (blockIdx.y * 8 + wave) * 16;
    const int m     = lane & 15;         // A row / B,C column within tile
    const int kh    = (lane >> 4) * 2;   // upper half-lanes hold K+2,K+3

    float bias = bih[rbase + m] + bhh[rbase + m];
    v8f acc;
    #pragma unroll
    for (int v = 0; v < 8; ++v) acc[v] = bias;

    const float* xrow = xs + m * EMB;                       // A: X[s0+m, :]
    const float* wrow = Wih + (size_t)(rbase + m) * EMB;    // B col n = W_ih row

    #pragma unroll 4
    for (int k0 = 0; k0 < EMB; k0 += 4) {
        v2f a, b;
        a.x = xrow[k0 + kh];
        a.y = xrow[k0 + kh + 1];
        b.x = wrow[k0 + kh];
        b.y = wrow[k0 + kh + 1];
        acc = __builtin_amdgcn_wmma_f32_16x16x4_f32(
                  false, a, false, b, (short)0, acc, false, false);
    }

    #pragma unroll
    for (int v = 0; v < 8; ++v) {
        int row = v + ((lane >> 4) << 3);  // C/D: VGPR v -> M = v or v+8
        gx[(size_t)(s0 + row) * G4 + rbase + m] = acc[v];
    }
}

// ---------------------------------------------------------------------------
// Phase 2: persistent cooperative scan. 16 WGs x 256 threads. WG `wg` owns
// hidden units [wg*32, wg*32+32) and their 4 gate rows (128 rows of W_hh,
// L2-resident). h is double-buffered in global memory; one device barrier
// per step. c lives in registers of threads 0..31 of the owning WG.
// ---------------------------------------------------------------------------
__global__ void __launch_bounds__(256)
lstm_scan(const float* __restrict__ gx, const float* __restrict__ Whh,
          float* __restrict__ hbuf, int* __restrict__ cnt,
          float* __restrict__ out) {
    __shared__ __align__(16) float h_s[HID];
    __shared__ float gate_s[128];

    const int tid  = threadIdx.x;
    const int wg   = blockIdx.x;      // 16 workgroups
    const int wave = tid >> 5;
    const int lane = tid & 31;
    const int NWG  = gridDim.x;

    float c_reg = 0.0f, h_last = 0.0f;

    for (int step = 0; step < SEQ; ++step) {
        const float* cur = hbuf + (step & 1) * HID;
        float*       nxt = hbuf + ((step + 1) & 1) * HID;

#ifdef USE_ASYNC_LDS
        // DMA h straight into LDS at device scope (L2-coherent vs other WGs).
        __builtin_amdgcn_global_load_async_to_lds_b32(
            (AS1 int*)(cur + tid), (AS3 int*)(h_s + tid), 0, CPOL_SCOPE_DEV);
        __builtin_amdgcn_global_load_async_to_lds_b32(
            (AS1 int*)(cur + tid + 256), (AS3 int*)(h_s + tid + 256), 0,
            CPOL_SCOPE_DEV);
        __builtin_amdgcn_s_wait_asynccnt(0);
        __syncthreads();
#else
        h_s[tid]       = __hip_atomic_load(cur + tid,       __ATOMIC_RELAXED,
                                           __HIP_MEMORY_SCOPE_AGENT);
        h_s[tid + 256] = __hip_atomic_load(cur + tid + 256, __ATOMIC_RELAXED,
                                           __HIP_MEMORY_SCOPE_AGENT);
        __syncthreads();
#endif

        const float4* h4 = (const float4*)h_s;
        // 16 gate rows per wave; lane-interleaved float4 dot + wave32 reduce.
        for (int i = 0; i < 16; ++i) {
            int lr   = wave * 16 + i;            // local row 0..127
            int gate = lr >> 5, uiw = lr & 31;   // gate-major layout
            int gr   = gate * HID + wg * 32 + uiw;
            const float4* w4 = (const float4*)(Whh + (size_t)gr * HID);
            float acc = 0.0f;
            #pragma unroll
            for (int k = 0; k < 4; ++k) {
                float4 w = w4[lane + 32 * k];
                float4 h = h4[lane + 32 * k];
                acc += w.x * h.x + w.y * h.y + w.z * h.z + w.w * h.w;
            }
            #pragma unroll
            for (int off = 16; off > 0; off >>= 1)
                acc += __shfl_xor(acc, off, 32);
            if (lane == 0) gate_s[lr] = acc;
        }
        __syncthreads();

        if (tid < 32) {
            const float* g0 = gx + (size_t)step * G4 + wg * 32 + tid;
            float zi = gate_s[tid]       + g0[0];
            float zf = gate_s[32 + tid]  + g0[HID];
            float zg = gate_s[64 + tid]  + g0[2 * HID];
            float zo = gate_s[96 + tid]  + g0[3 * HID];
            float ii = 1.0f / (1.0f + __expf(-zi));
            float ff = 1.0f / (1.0f + __expf(-zf));
            float gg = tanhf(zg);
            float oo = 1.0f / (1.0f + __expf(-zo));
            c_reg  = ff * c_reg + ii * gg;
            h_last = oo * tanhf(c_reg);
            __hip_atomic_store(nxt + wg * 32 + tid, h_last, __ATOMIC_RELAXED,
                               __HIP_MEMORY_SCOPE_AGENT);
            if (step + 1 < SEQ)  // warm L2->WGP path for next step's gx row
                __builtin_prefetch(gx + (size_t)(step + 1) * G4 + wg * 32 + tid, 0, 1);
        }
        __threadfence();
        __syncthreads();
        if (tid == 0) {
            __hip_atomic_fetch_add(cnt, 1, __ATOMIC_RELEASE,
                                   __HIP_MEMORY_SCOPE_AGENT);
            int target = NWG * (step + 1);
            while (__hip_atomic_load(cnt, __ATOMIC_ACQUIRE,
                                     __HIP_MEMORY_SCOPE_AGENT) < target)
                __builtin_amdgcn_s_sleep(2);
        }
        __syncthreads();
    }

    if (tid < 32) {
        out[wg * 32 + tid]       = h_last;  // h
        out[HID + wg * 32 + tid] = c_reg;   // c
    }
}

// ---------------------------------------------------------------------------
extern "C" void kernel_launch(void* const* d_in, const int* in_sizes, int n_in,
                              void* d_out, int out_size, void* d_ws, size_t ws_size,
                              hipStream_t stream) {
    const int*   tokens = (const int*)d_in[0];
    const float* emb    = (const float*)d_in[1];
    const float* Wih    = (const float*)d_in[2];
    const float* Whh    = (const float*)d_in[3];
    const float* bih    = (const float*)d_in[4];
    const float* bhh    = (const float*)d_in[5];
    float* out = (float*)d_out;

    // workspace: gx [SEQ*G4] | h double-buffer [2*HID] | barrier counter
    float* gx   = (float*)d_ws;
    float* hbuf = gx + (size_t)SEQ * G4;
    int*   cnt  = (int*)(hbuf + 2 * HID);

    init_state<<<1, 256, 0, stream>>>(hbuf, cnt);

    dim3 g1(SEQ / 16, G4 / (16 * 8));  // 256 s-tiles x 16 blocks of 8 r-tiles
    gx_gemm<<<g1, 256, 0, stream>>>(tokens, emb, Wih, bih, bhh, gx);

    lstm_scan<<<16, 256, 0, stream>>>(gx, Whh, hbuf, cnt, out);
}